// MultiResFeatureGrid2D_59837484367919
// MI455X (gfx1250) — compile-verified
//
#include <hip/hip_runtime.h>
#include <hip/hip_fp16.h>

#define BLOCK 256
#define PPT   16
#define NLVL  8
#define NLDS  3            // levels staged in LDS: 16^2, 32^2, 64^2
#define LDS_ELEMS 5376     // float2 elements: 256 + 1024 + 4096  (43 KB)

// Async global -> LDS copy of nElem float2 (8B) elements, block-cooperative,
// in 16B chunks. nElem is always a multiple of 2 here; base pointers are
// 16B-aligned (hipMalloc is 256B-aligned, LDS offsets are 16B-aligned).
// Uses CDNA5 GLOBAL_LOAD_ASYNC_TO_LDS_B128 (ASYNCcnt path) via inline asm:
// the clang builtin's pointer params are in the cuda_device/cuda_shared lang
// address spaces which cannot be spelled in a HIP cast.
__device__ __forceinline__ void stage_level(const float2* __restrict__ g,
                                            float2* lds_dst, int nElem, int tid)
{
    const int nVec = nElem >> 1;
    for (int i = tid; i < nVec; i += BLOCK) {
        const unsigned long long ga = (unsigned long long)(const void*)(g + 2 * i);
        // low 32 bits of a flat shared pointer == LDS byte offset (ISA 10.2 aperture rule)
        const unsigned la = (unsigned)(unsigned long long)(void*)(lds_dst + 2 * i);
        asm volatile("global_load_async_to_lds_b128 %0, %1, off"
                     :: "v"(la), "v"(ga)
                     : "memory");
    }
}

__global__ __launch_bounds__(BLOCK) void mrgrid2d_kernel(
    const float2* __restrict__ coords,
    const float2* __restrict__ g0, const float2* __restrict__ g1,
    const float2* __restrict__ g2, const float2* __restrict__ g3,
    const float2* __restrict__ g4, const float2* __restrict__ g5,
    const float2* __restrict__ g6, const float2* __restrict__ g7,
    float2* __restrict__ out, int n)
{
    constexpr int RES[NLVL]  = {16, 32, 64, 128, 256, 512, 1024, 2048};
    constexpr int LOFF[NLDS] = {0, 256, 1280};   // float2 offsets in LDS

    __shared__ float2 smem[LDS_ELEMS];

    const int tid = threadIdx.x;
    stage_level(g0, smem + LOFF[0], 256,  tid);
    stage_level(g1, smem + LOFF[1], 1024, tid);
    stage_level(g2, smem + LOFF[2], 4096, tid);
    // wait for this wave's outstanding async copies, then publish to the block
    asm volatile("s_wait_asynccnt 0" ::: "memory");
    __syncthreads();

    const float2* __restrict__ gg[NLVL - NLDS] = {g3, g4, g5, g6, g7};

    const int i0 = (int)blockIdx.x * (BLOCK * PPT) + tid;
    for (int k = 0; k < PPT; ++k) {
        const int i = i0 + k * BLOCK;
        if (i >= n) return;   // no barriers after this point; stride-BLOCK so monotone

        const float2 c = coords[i];
        const float x = fminf(fmaxf(c.x, 0.0f), 1.0f - 1e-6f);
        const float y = fminf(fmaxf(c.y, 0.0f), 1.0f - 1e-6f);

        // ---- phase 1: compute all addresses, issue all 32 gathers ----
        float2 f00[NLVL], f10[NLVL], f01[NLVL], f11[NLVL];
        float  fx[NLVL], fy[NLVL];
#pragma unroll
        for (int l = 0; l < NLVL; ++l) {
            const int   r  = RES[l];
            const float s  = (float)(r - 1);
            const float xs = x * s;
            const float ys = y * s;
            int x0 = (int)xs;  if (x0 > r - 2) x0 = r - 2;   // xs >= 0 -> trunc==floor
            int y0 = (int)ys;  if (y0 > r - 2) y0 = r - 2;
            fx[l] = xs - (float)x0;
            fy[l] = ys - (float)y0;
            const int base = y0 * r + x0;
            if (l < NLDS) {
                const float2* p = smem + LOFF[l];
                f00[l] = p[base];     f10[l] = p[base + 1];
                f01[l] = p[base + r]; f11[l] = p[base + r + 1];
            } else {
                const float2* __restrict__ p = gg[l - NLDS];
                f00[l] = p[base];     f10[l] = p[base + 1];
                f01[l] = p[base + r]; f11[l] = p[base + r + 1];
            }
        }

        // ---- phase 2: bilinear in f32, round to fp16 per level, fp16 accumulate ----
        __half2 acc = __float2half2_rn(0.0f);
#pragma unroll
        for (int l = 0; l < NLVL; ++l) {
            const float ox = 1.0f - fx[l];
            const float oy = 1.0f - fy[l];
            const float r0x = f00[l].x * ox + f10[l].x * fx[l];
            const float r0y = f00[l].y * ox + f10[l].y * fx[l];
            const float r1x = f01[l].x * ox + f11[l].x * fx[l];
            const float r1y = f01[l].y * ox + f11[l].y * fx[l];
            const float vx  = r0x * oy + r1x * fy[l];
            const float vy  = r0y * oy + r1y * fy[l];
            acc = __hadd2(acc, __floats2half2_rn(vx, vy));
        }

        float2 o;
        o.x = __low2float(acc);
        o.y = __high2float(acc);
        out[i] = o;
    }
}

extern "C" void kernel_launch(void* const* d_in, const int* in_sizes, int n_in,
                              void* d_out, int out_size, void* d_ws, size_t ws_size,
                              hipStream_t stream) {
    const float2* coords = (const float2*)d_in[0];
    const float2* g0 = (const float2*)d_in[1];
    const float2* g1 = (const float2*)d_in[2];
    const float2* g2 = (const float2*)d_in[3];
    const float2* g3 = (const float2*)d_in[4];
    const float2* g4 = (const float2*)d_in[5];
    const float2* g5 = (const float2*)d_in[6];
    const float2* g6 = (const float2*)d_in[7];
    const float2* g7 = (const float2*)d_in[8];
    float2* out = (float2*)d_out;

    const int n = in_sizes[0] / 2;                    // number of points
    const int blocks = (n + BLOCK * PPT - 1) / (BLOCK * PPT);
    mrgrid2d_kernel<<<blocks, BLOCK, 0, stream>>>(coords, g0, g1, g2, g3, g4,
                                                  g5, g6, g7, out, n);
}